// TSNE_17583596110566
// MI455X (gfx1250) — compile-verified
//
#include <hip/hip_runtime.h>
#include <math.h>

// ext-vector types for the WMMA operands:
//   A: 16x4 f32  -> 2 VGPRs,  B: 4x16 f32 -> 2 VGPRs,  C/D: 16x16 f32 -> 8 VGPRs
typedef __attribute__((ext_vector_type(2))) float v2f;
typedef __attribute__((ext_vector_type(8))) float v8f;

#define NCHUNK 8  // column-loop split for occupancy (grid.y)

// ---------------------------------------------------------------------------
// Kernel 1: partition term via V_WMMA_F32_16X16X4_F32.
// One wave owns a 16-row strip and one column chunk.
// A row m  = [sq_m, 1, -2x_m, -2y_m], B col n = [1, sq_n, x_n, y_n]
//   => D[m][n] = sq_m + sq_n - 2*(x_m x_n + y_m y_n) = d^2(m,n)
// ---------------------------------------------------------------------------
__global__ void tsne_part_wmma(const float* __restrict__ logits,
                               float* __restrict__ wsA, int n, int ntiles,
                               int tilesPerChunk) {
  extern __shared__ float2 sxy[];  // n entries = 80 KB for n = 10000
  const int tid = threadIdx.x;
  for (int idx = tid; idx < n; idx += blockDim.x)
    sxy[idx] = make_float2(logits[2 * idx], logits[2 * idx + 1]);
  __syncthreads();

  const int lane = tid & 31;
  const int wave = tid >> 5;
  const int wavesPerBlock = blockDim.x >> 5;
  const int rowTile = blockIdx.x * wavesPerBlock + wave;  // wave-uniform

  float acc0 = 0.0f, acc1 = 0.0f, acc2 = 0.0f, acc3 = 0.0f;
  if (rowTile < ntiles) {                 // wave-uniform: EXEC stays all-ones
    const int m = rowTile * 16 + (lane & 15);  // n is a multiple of 16
    const float2 pm = sxy[m];
    const float sqm = pm.x * pm.x + pm.y * pm.y;
    // A-matrix 16x4 layout: lanes 0-15 hold K=0,1; lanes 16-31 hold K=2,3
    v2f a;
    a.x = (lane < 16) ? sqm  : (-2.0f * pm.x);
    a.y = (lane < 16) ? 1.0f : (-2.0f * pm.y);

    const int c0 = blockIdx.y * tilesPerChunk;
    const int c1 = min(ntiles, c0 + tilesPerChunk);
    for (int ct = c0; ct < c1; ++ct) {
      const int nn = ct * 16 + (lane & 15);
      const float2 pn = sxy[nn];
      const float sqn = pn.x * pn.x + pn.y * pn.y;
      // B-matrix 4x16 layout mirrors A: lanes 0-15 K=0,1; lanes 16-31 K=2,3
      v2f b;
      b.x = (lane < 16) ? 1.0f : pn.x;
      b.y = (lane < 16) ? sqn  : pn.y;

      v8f c = {};
      // (neg_a, A, neg_b, B, c_mod, C, reuse_a, reuse_b)
      c = __builtin_amdgcn_wmma_f32_16x16x4_f32(false, a, false, b,
                                                (short)0, c, false, false);
#pragma unroll
      for (int v = 0; v < 8; ++v) {
        // single v_med3_f32 clamp (no canonicalize, unlike fmaxf)
        const float d2 = __builtin_amdgcn_fmed3f(c[v], 0.0f, 3.0e38f);
        const float x = 1.0f + d2;
        float r = __builtin_amdgcn_rcpf(x);
        r = r * (2.0f - x * r);  // one Newton step -> ~f32-div accuracy
        if ((v & 3) == 0)      acc0 += r;
        else if ((v & 3) == 1) acc1 += r;
        else if ((v & 3) == 2) acc2 += r;
        else                   acc3 += r;
      }
    }
  }
  float acc = (acc0 + acc1) + (acc2 + acc3);
  // wave32 reduction; every wave writes its slot (deterministic, no init pass)
  for (int off = 16; off > 0; off >>= 1) acc += __shfl_down(acc, off, 32);
  if (lane == 0) {
    const int slot = blockIdx.y * (gridDim.x * wavesPerBlock) +
                     blockIdx.x * wavesPerBlock + wave;
    wsA[slot] = (rowTile < ntiles) ? acc : 0.0f;
  }
}

// ---------------------------------------------------------------------------
// Kernel 2: sampled-pair term.  KL = sum p*(log p + log(1+d2)) + log(part)*sum p
// ---------------------------------------------------------------------------
__global__ void tsne_pair(const float* __restrict__ pij,
                          const int* __restrict__ ii,
                          const int* __restrict__ jj,
                          const float* __restrict__ logits,
                          float* __restrict__ wsB1, float* __restrict__ wsB2,
                          int npairs) {
  const int tid = threadIdx.x;
  const int stride = gridDim.x * blockDim.x;
  float a1 = 0.0f, a2 = 0.0f;
  for (int k = blockIdx.x * blockDim.x + tid; k < npairs; k += stride) {
    const float p = pij[k];
    const int ia = ii[k], ja = jj[k];
    const float xa = logits[2 * ia], ya = logits[2 * ia + 1];
    const float xb = logits[2 * ja], yb = logits[2 * ja + 1];
    const float dx = xa - xb, dy = ya - yb;
    const float d2 = dx * dx + dy * dy;
    a1 += p * (logf(p) + logf(1.0f + d2));  // p*(log p - log num)
    a2 += p;
  }
  const int lane = tid & 31, wave = tid >> 5;
  __shared__ float s1[8], s2[8];
  for (int off = 16; off > 0; off >>= 1) {
    a1 += __shfl_down(a1, off, 32);
    a2 += __shfl_down(a2, off, 32);
  }
  if (lane == 0) { s1[wave] = a1; s2[wave] = a2; }
  __syncthreads();
  if (tid == 0) {
    float t1 = 0.0f, t2 = 0.0f;
    const int nw = (int)(blockDim.x >> 5);
    for (int w = 0; w < nw; ++w) { t1 += s1[w]; t2 += s2[w]; }
    wsB1[blockIdx.x] = t1;
    wsB2[blockIdx.x] = t2;
  }
}

// ---------------------------------------------------------------------------
// Kernel 3: deterministic final reduction + scalar combine.
// ---------------------------------------------------------------------------
__global__ void tsne_finalize(const float* __restrict__ wsA, int slotsA,
                              const float* __restrict__ wsB1,
                              const float* __restrict__ wsB2, int nb, int n,
                              float* __restrict__ out) {
  __shared__ float red[256];
  const int tid = threadIdx.x;

  float a = 0.0f;
  for (int k = tid; k < slotsA; k += 256) a += wsA[k];
  red[tid] = a; __syncthreads();
  for (int s = 128; s > 0; s >>= 1) { if (tid < s) red[tid] += red[tid + s]; __syncthreads(); }
  const float part = red[0] - (float)n;   // exclude diagonal (n ones)
  __syncthreads();

  float b1 = 0.0f;
  for (int k = tid; k < nb; k += 256) b1 += wsB1[k];
  red[tid] = b1; __syncthreads();
  for (int s = 128; s > 0; s >>= 1) { if (tid < s) red[tid] += red[tid + s]; __syncthreads(); }
  const float sum1 = red[0];
  __syncthreads();

  float b2 = 0.0f;
  for (int k = tid; k < nb; k += 256) b2 += wsB2[k];
  red[tid] = b2; __syncthreads();
  for (int s = 128; s > 0; s >>= 1) { if (tid < s) red[tid] += red[tid + s]; __syncthreads(); }

  if (tid == 0) out[0] = sum1 + logf(part) * red[0];
}

// ---------------------------------------------------------------------------
extern "C" void kernel_launch(void* const* d_in, const int* in_sizes, int n_in,
                              void* d_out, int out_size, void* d_ws, size_t ws_size,
                              hipStream_t stream) {
  const float* pij    = (const float*)d_in[0];
  const int*   ii     = (const int*)d_in[1];
  const int*   jj     = (const int*)d_in[2];
  const float* logits = (const float*)d_in[3];

  const int npairs = in_sizes[0];
  const int n      = in_sizes[3] / 2;        // N_POINTS (D = 2)
  const int ntiles = (n + 15) / 16;          // 625 row tiles for n = 10000
  const int blocksA = (ntiles + 7) / 8;      // 8 waves per 256-thread block
  const int tilesPerChunk = (ntiles + NCHUNK - 1) / NCHUNK;
  const int slotsA  = blocksA * 8 * NCHUNK;  // per-wave partials, all chunks
  const int NB = 1024;                       // pair-kernel blocks

  float* ws   = (float*)d_ws;
  float* wsA  = ws;                          // [slotsA]
  float* wsB1 = ws + slotsA;                 // [NB]
  float* wsB2 = wsB1 + NB;                   // [NB]

  const size_t ldsBytes = (size_t)n * sizeof(float2);  // 80 KB < 320 KB/WGP

  dim3 gridA(blocksA, NCHUNK);
  tsne_part_wmma<<<gridA, 256, ldsBytes, stream>>>(logits, wsA, n, ntiles,
                                                   tilesPerChunk);
  tsne_pair<<<NB, 256, 0, stream>>>(pij, ii, jj, logits, wsB1, wsB2, npairs);
  tsne_finalize<<<1, 256, 0, stream>>>(wsA, slotsA, wsB1, wsB2, NB, n,
                                       (float*)d_out);
}